// RejectionSampler_28613072126249
// MI455X (gfx1250) — compile-verified
//
#include <hip/hip_runtime.h>

#define VOCAB 128000
#define BATCH 64
#define NSPEC 8
#define ROWS (BATCH * NSPEC)        // 512
#define PARTS 5                     // blocks per row
#define THREADS 256
#define F4_PER_ROW (VOCAB / 4)      // 32000
#define F4_PER_PART (F4_PER_ROW / PARTS)   // 6400
#define ITERS (F4_PER_PART / THREADS)      // 25

typedef float f4 __attribute__((ext_vector_type(4)));

// ---------------------------------------------------------------------------
// Kernel 1: partial argmax over a 1/PARTS slice of one vocab row.
// Streaming, read-once: non-temporal b128 loads (262MB working set > 192MB
// L2, so don't pollute it). Inner loop tracks (group_max, group_idx) only
// (~1.2 VALU/element) so VALU stays well under the HBM roofline; the winning
// component inside the 4-wide group is resolved once per block at the end.
// Strict '>' + ascending traversal + index tie-break == jnp.argmax
// first-occurrence semantics.
// ---------------------------------------------------------------------------
__global__ __launch_bounds__(THREADS) void partial_argmax_kernel(
    const float* __restrict__ logits,
    float* __restrict__ ws_val,
    int* __restrict__ ws_idx) {
  const int row  = blockIdx.x / PARTS;   // 0..511
  const int part = blockIdx.x % PARTS;   // 0..4
  const int tid  = threadIdx.x;

  const f4* base = (const f4*)(logits + (size_t)row * VOCAB);
  const int part_base = part * F4_PER_PART;

  float maxv = -__builtin_inff();
  int   maxg = part_base;                // winning float4-group index

#pragma unroll 5
  for (int it = 0; it < ITERS; ++it) {
    const int idx4 = part_base + it * THREADS + tid;   // ascending per thread
    const f4 v = __builtin_nontemporal_load(base + idx4);
    const float gm = __builtin_fmaxf(__builtin_fmaxf(v.x, v.y),
                                     __builtin_fmaxf(v.z, v.w));
    if (gm > maxv) { maxv = gm; maxg = idx4; }
  }

  // wave32 butterfly reduction; lexicographic (val desc, group idx asc)
#pragma unroll
  for (int off = 16; off > 0; off >>= 1) {
    const float ov = __shfl_xor(maxv, off, 32);
    const int   og = __shfl_xor(maxg, off, 32);
    if (ov > maxv || (ov == maxv && og < maxg)) { maxv = ov; maxg = og; }
  }

  __shared__ float sv[THREADS / 32];
  __shared__ int   sg[THREADS / 32];
  const int wave = tid >> 5;
  const int lane = tid & 31;
  if (lane == 0) { sv[wave] = maxv; sg[wave] = maxg; }
  __syncthreads();

  if (tid == 0) {
    float bv = sv[0];
    int   bg = sg[0];
#pragma unroll
    for (int w = 1; w < THREADS / 32; ++w) {
      if (sv[w] > bv || (sv[w] == bv && sg[w] < bg)) { bv = sv[w]; bg = sg[w]; }
    }
    // Resolve the component within the winning group: first equal wins.
    const f4 g = __builtin_nontemporal_load(base + bg);
    int c = 3;
    if (g.z == bv) c = 2;
    if (g.y == bv) c = 1;
    if (g.x == bv) c = 0;
    ws_val[blockIdx.x] = bv;
    ws_idx[blockIdx.x] = bg * 4 + c;
  }
}

// ---------------------------------------------------------------------------
// Kernel 2: combine partials -> target tokens, then rejection-sampling logic.
// One thread per batch row. Writes all 704 floats of d_out:
//   [0,576)   output_token_ids  (B x 9)
//   [576,640) num_rejected_tokens (B)
//   [640,704) last_token_ids (B)
// ---------------------------------------------------------------------------
__global__ __launch_bounds__(64) void rejection_finalize_kernel(
    const float* __restrict__ ws_val,
    const int* __restrict__ ws_idx,
    const int* __restrict__ draft,    // [B, NSPEC]
    const int* __restrict__ bonus,    // [B, 1]
    float* __restrict__ out) {
  const int b = threadIdx.x;
  if (b >= BATCH) return;

  // 1) combine the PARTS partials per spec position -> argmax token ids
  int tgt[NSPEC];
#pragma unroll
  for (int j = 0; j < NSPEC; ++j) {
    const int row = b * NSPEC + j;
    float bv = ws_val[row * PARTS];
    int   bi = ws_idx[row * PARTS];
#pragma unroll
    for (int p = 1; p < PARTS; ++p) {
      const float v = ws_val[row * PARTS + p];
      const int   i = ws_idx[row * PARTS + p];
      if (v > bv || (v == bv && i < bi)) { bv = v; bi = i; }
    }
    tgt[j] = bi;
  }

  // 2) length of accepted prefix: equals[j] == (j < n)
  int n = 0;
  bool run = true;
#pragma unroll
  for (int j = 0; j < NSPEC; ++j) {
    const bool m = (draft[b * NSPEC + j] == tgt[j]);
    run = run && m;
    n += run ? 1 : 0;
  }

  const int num_rejected = NSPEC - n;
  // jnp.argmin(equals): first False index, or 0 if all True
  const int first_diff = (n == NSPEC) ? 0 : n;

  float outrow[NSPEC + 1];
  int cnt = 0;
#pragma unroll
  for (int j = 0; j < NSPEC; ++j) {
    const bool keep = (j <= first_diff) || (j < n);   // keeps | equals
    outrow[j] = keep ? (float)tgt[j] : -1.0f;
    cnt += keep ? 1 : 0;
  }
  const bool bonus_keep = (n == NSPEC);
  outrow[NSPEC] = bonus_keep ? (float)bonus[b] : -1.0f;
  cnt += bonus_keep ? 1 : 0;

#pragma unroll
  for (int j = 0; j <= NSPEC; ++j)
    out[b * (NSPEC + 1) + j] = outrow[j];
  out[BATCH * (NSPEC + 1) + b] = (float)num_rejected;
  out[BATCH * (NSPEC + 1) + BATCH + b] = outrow[cnt - 1];
}

extern "C" void kernel_launch(void* const* d_in, const int* in_sizes, int n_in,
                              void* d_out, int out_size, void* d_ws, size_t ws_size,
                              hipStream_t stream) {
  (void)in_sizes; (void)n_in; (void)out_size; (void)ws_size;

  const float* logits = (const float*)d_in[0];   // [64, 8, 128000] f32
  const int*   draft  = (const int*)d_in[1];     // [64, 8]
  const int*   bonus  = (const int*)d_in[2];     // [64, 1]

  // workspace layout: PARTS*ROWS floats then PARTS*ROWS ints (20 KB total)
  float* ws_val = (float*)d_ws;
  int*   ws_idx = (int*)(ws_val + ROWS * PARTS);

  partial_argmax_kernel<<<ROWS * PARTS, THREADS, 0, stream>>>(logits, ws_val, ws_idx);
  rejection_finalize_kernel<<<1, 64, 0, stream>>>(ws_val, ws_idx, draft, bonus,
                                                  (float*)d_out);
}